// MultiBoxLoss_61125974557237
// MI455X (gfx1250) — compile-verified
//
#include <hip/hip_runtime.h>
#include <stdint.h>

// Problem constants (from reference)
#define NB 64      // batches
#define NT 50      // objects (truths) per batch
#define ND 8732    // default boxes
#define THREADS 256

typedef float v2f __attribute__((ext_vector_type(2)));
typedef float v8f __attribute__((ext_vector_type(8)));

__device__ __forceinline__ float smooth_l1_term(float d) {
    float a = fabsf(d);
    return (a < 1.0f) ? 0.5f * a * a : a - 0.5f;
}

// ---------------------------------------------------------------------------
// Kernel 1: per-batch match + encode + smooth-L1 + CE, one block per batch.
//   - truths tile pulled into LDS via the Tensor Data Mover (gfx1250 TDM)
//   - per-truth argmax over 8732 priors via packed u64 LDS atomicMax
//   - per-prior best kept in LDS (52KB table, WGP has 320KB)
//   - writes mined CE array + per-batch partial sums (no global atomics)
// ---------------------------------------------------------------------------
__global__ __launch_bounds__(THREADS) void k_match(
    const float* __restrict__ loc_data,   // [NB][ND][4]
    const float* __restrict__ conf_data,  // [NB][ND][2]
    const float* __restrict__ dbox,       // [ND][4] cx,cy,w,h
    const float* __restrict__ targets,    // [NB][NT][5]
    float* __restrict__ mined,            // [NB][ND] ce with 0 at pos
    int*   __restrict__ pb_npos,          // [NB]
    float* __restrict__ pb_posf,          // [NB]
    float* __restrict__ pb_l1,            // [NB]
    float* __restrict__ pb_cep)           // [NB]
{
    // s_tgt MUST be the first shared object: TDM descriptor uses lds_addr = 0.
    __shared__ float s_tgt[NT * 5];
    __shared__ float s_area[NT];
    __shared__ unsigned long long s_bestT[NT];   // (iou_bits<<32)|~prior_idx
    __shared__ float s_ov[ND];                   // best overlap per prior
    __shared__ unsigned short s_ti[ND];          // best truth idx per prior
    __shared__ float s_red[3][THREADS / 32];

    const int b   = blockIdx.x;
    const int tid = threadIdx.x;

    // --- Stage targets[b] (50x5 f32, contiguous 250 floats) into LDS ---
#if __has_builtin(__builtin_amdgcn_tensor_load_to_lds) && __has_builtin(__builtin_amdgcn_s_wait_tensorcnt)
    if (tid < 32) {  // one wave issues the DMA (EXEC ignored by TDM)
        typedef unsigned int u32x4 __attribute__((ext_vector_type(4)));
        typedef int          i32x4 __attribute__((ext_vector_type(4)));
        typedef int          i32x8 __attribute__((ext_vector_type(8)));
        unsigned long long ga =
            (unsigned long long)(const void*)(targets + (size_t)b * NT * 5);
        u32x4 g0;
        g0[0] = 1u;                                   // count=1 (valid D#)
        g0[1] = 0u;                                   // lds_addr = 0 (s_tgt)
        g0[2] = (unsigned)(ga & 0xFFFFFFFFu);         // global_addr[31:0]
        g0[3] = (unsigned)((ga >> 32) & 0x1FFFFFFu)   // global_addr[56:32]
              | (2u << 30);                           // type = 2 ("image")
        i32x8 g1;
        g1[0] = (int)(2u << 16);                      // data_size = 4 bytes
        g1[1] = (int)(5u << 16);                      // tensor_dim0 = 5
        g1[2] = (int)((unsigned)NT << 16);            // dim0 hi | tensor_dim1 = 50
        g1[3] = (int)(5u << 16);                      // dim1 hi | tile_dim0 = 5
        g1[4] = (int)NT;                              // tile_dim1 = 50, tile_dim2 = 0
        g1[5] = 5;                                    // tensor_dim0_stride = 5
        g1[6] = 0;
        g1[7] = 0;
        i32x4 g2 = {0, 0, 0, 0};                      // no dims 2..4
        i32x4 g3 = {0, 0, 0, 0};
        i32x8 g4 = {0, 0, 0, 0, 0, 0, 0, 0};          // trailing group (clang-23 6-arg form)
        __builtin_amdgcn_tensor_load_to_lds(g0, g1, g2, g3, g4, 0);
        __builtin_amdgcn_s_wait_tensorcnt(0);
    }
#else
    for (int i = tid; i < NT * 5; i += blockDim.x)
        s_tgt[i] = targets[(size_t)b * NT * 5 + i];
#endif
    // gfx1250 cacheline prefetch of this batch's logits/locs
    __builtin_prefetch(conf_data + ((size_t)b * ND) * 2 + tid * 32, 0, 1);
    __builtin_prefetch(loc_data  + ((size_t)b * ND) * 4 + tid * 32, 0, 1);

    for (int t = tid; t < NT; t += blockDim.x) {
        s_bestT[t] = 0ull;
    }
    __syncthreads();   // s_tgt visible to all waves from here
    for (int t = tid; t < NT; t += blockDim.x) {
        float x0 = s_tgt[t * 5 + 0], y0 = s_tgt[t * 5 + 1];
        float x1 = s_tgt[t * 5 + 2], y1 = s_tgt[t * 5 + 3];
        s_area[t] = (x1 - x0) * (y1 - y0);
    }
    __syncthreads();

    // --- Pass 1: IoU scan; per-prior argmax (regs) + per-truth argmax (LDS) ---
    for (int j = tid; j < ND; j += blockDim.x) {
        float cx = dbox[j * 4 + 0], cy = dbox[j * 4 + 1];
        float w  = dbox[j * 4 + 2], h  = dbox[j * 4 + 3];
        float dx0 = cx - 0.5f * w, dy0 = cy - 0.5f * h;
        float dx1 = cx + 0.5f * w, dy1 = cy + 0.5f * h;
        float ad  = w * h;
        float bov = -1.0f;
        int   bt  = 0;
        for (int t = 0; t < NT; ++t) {
            float tx0 = s_tgt[t * 5 + 0], ty0 = s_tgt[t * 5 + 1];
            float tx1 = s_tgt[t * 5 + 2], ty1 = s_tgt[t * 5 + 3];
            float iw = fmaxf(fminf(dx1, tx1) - fmaxf(dx0, tx0), 0.0f);
            float ih = fmaxf(fminf(dy1, ty1) - fmaxf(dy0, ty0), 0.0f);
            float inter = iw * ih;
            float iou = inter / (s_area[t] + ad - inter);
            if (iou > bov) { bov = iou; bt = t; }   // first-occurrence argmax
            // per-truth max over priors; iou>=0 so f32 bits are monotonic;
            // ~j breaks ties toward smaller prior index (argmax semantics)
            unsigned long long p =
                ((unsigned long long)__float_as_uint(iou) << 32) |
                (unsigned)(~(unsigned)j);
            if (p > s_bestT[t]) atomicMax(&s_bestT[t], p);
        }
        s_ov[j] = bov;
        s_ti[j] = (unsigned short)bt;
    }
    __syncthreads();

    // --- best_prior override: ov=2.0, truth idx forced (serial = last wins) ---
    if (tid == 0) {
        for (int t = 0; t < NT; ++t) {
            unsigned jstar = ~(unsigned)(s_bestT[t] & 0xFFFFFFFFull);
            s_ov[jstar] = 2.0f;
            s_ti[jstar] = (unsigned short)t;
        }
    }
    __syncthreads();

    // --- Pass 2: encode, smooth-L1, cross-entropy, mined CE ---
    float a_l1 = 0.0f, a_cep = 0.0f;
    int   a_np = 0;
    for (int j = tid; j < ND; j += blockDim.x) {
        float ov  = s_ov[j];
        int   t   = s_ti[j];
        bool  pos = (ov >= 0.5f);
        float c0 = conf_data[((size_t)b * ND + j) * 2 + 0];
        float c1 = conf_data[((size_t)b * ND + j) * 2 + 1];
        float m   = fmaxf(c0, c1);
        float lse = m + logf(expf(c0 - m) + expf(c1 - m));
        float ce  = lse - (pos ? c1 : c0);   // label = conf_t (0 or 1)
        mined[(size_t)b * ND + j] = pos ? 0.0f : ce;
        if (pos) {
            a_cep += ce;
            a_np  += 1;
            float tx0 = s_tgt[t * 5 + 0], ty0 = s_tgt[t * 5 + 1];
            float tx1 = s_tgt[t * 5 + 2], ty1 = s_tgt[t * 5 + 3];
            float cx = dbox[j * 4 + 0], cy = dbox[j * 4 + 1];
            float w  = dbox[j * 4 + 2], h  = dbox[j * 4 + 3];
            float gcx = (0.5f * (tx0 + tx1) - cx) / (0.1f * w);
            float gcy = (0.5f * (ty0 + ty1) - cy) / (0.1f * h);
            float gw  = logf((tx1 - tx0) / w) / 0.2f;
            float gh  = logf((ty1 - ty0) / h) / 0.2f;
            const float* L = &loc_data[((size_t)b * ND + j) * 4];
            a_l1 += smooth_l1_term(L[0] - gcx) + smooth_l1_term(L[1] - gcy) +
                    smooth_l1_term(L[2] - gw)  + smooth_l1_term(L[3] - gh);
        }
    }

    // --- deterministic block reduction (wave32 shuffles + LDS) ---
    float v0 = (float)a_np, v1 = a_l1, v2 = a_cep;
    for (int mlane = 16; mlane >= 1; mlane >>= 1) {
        v0 += __shfl_xor(v0, mlane, 32);
        v1 += __shfl_xor(v1, mlane, 32);
        v2 += __shfl_xor(v2, mlane, 32);
    }
    int wid = tid >> 5, lid = tid & 31;
    if (lid == 0) { s_red[0][wid] = v0; s_red[1][wid] = v1; s_red[2][wid] = v2; }
    __syncthreads();
    if (tid == 0) {
        float t0 = 0.0f, t1 = 0.0f, t2 = 0.0f;
        for (int wv = 0; wv < THREADS / 32; ++wv) {
            t0 += s_red[0][wv]; t1 += s_red[1][wv]; t2 += s_red[2][wv];
        }
        pb_npos[b] = (int)t0;
        pb_posf[b] = t0;
        pb_l1[b]   = t1;
        pb_cep[b]  = t2;
    }
}

// ---------------------------------------------------------------------------
// Kernel 2: per-batch hard-negative mining via exact 32-bit radix select.
// Sum of top-K is tie-invariant: sum(> vK) + (K - count_gt) * vK.
// ---------------------------------------------------------------------------
__global__ __launch_bounds__(THREADS) void k_mine(
    const float* __restrict__ mined_all,  // [NB][ND]
    const int*   __restrict__ pb_npos,    // [NB]
    float* __restrict__ pb_negsum,        // [NB]
    float* __restrict__ pb_negcnt)        // [NB]
{
    __shared__ int      hist[256];
    __shared__ unsigned s_prefix;
    __shared__ int      s_krem;
    __shared__ float    s_redf[THREADS / 32];

    const int b   = blockIdx.x;
    const int tid = threadIdx.x;
    const float* mined = mined_all + (size_t)b * ND;

    int K = pb_npos[b] * 3;
    K = (K < 20) ? 20 : K;
    K = (K > ND) ? ND : K;

    if (tid == 0) { s_prefix = 0u; s_krem = K; }
    __syncthreads();

    for (int round = 0; round < 4; ++round) {
        const int shift = 24 - 8 * round;
        for (int i = tid; i < 256; i += blockDim.x) hist[i] = 0;
        __syncthreads();
        unsigned pref = s_prefix;
        for (int j = tid; j < ND; j += blockDim.x) {
            unsigned u = __float_as_uint(mined[j]);   // all values >= 0
            if (round == 0 || (u >> (shift + 8)) == pref)
                atomicAdd(&hist[(u >> shift) & 0xFFu], 1);
        }
        __syncthreads();
        if (tid == 0) {
            int krem = s_krem;
            int bin = 255;
            while (bin > 0 && krem > hist[bin]) { krem -= hist[bin]; --bin; }
            s_prefix = (pref << 8) | (unsigned)bin;
            s_krem   = krem;
        }
        __syncthreads();
    }

    const unsigned vK_bits = s_prefix;     // exact bit pattern of K-th largest
    const float    vK      = __uint_as_float(vK_bits);
    const int      krem    = s_krem;       // # of ties at vK to include

    float sgt = 0.0f;
    for (int j = tid; j < ND; j += blockDim.x) {
        float v = mined[j];
        if (__float_as_uint(v) > vK_bits) sgt += v;
    }
    for (int mlane = 16; mlane >= 1; mlane >>= 1) sgt += __shfl_xor(sgt, mlane, 32);
    int wid = tid >> 5, lid = tid & 31;
    if (lid == 0) s_redf[wid] = sgt;
    __syncthreads();
    if (tid == 0) {
        float tot = 0.0f;
        for (int wv = 0; wv < THREADS / 32; ++wv) tot += s_redf[wv];
        pb_negsum[b] = tot + (float)krem * vK;
        pb_negcnt[b] = (float)K;
    }
}

// ---------------------------------------------------------------------------
// Kernel 3: finalize. 64-element reductions via v_wmma_f32_16x16x4_f32:
// A = ones(16x4), B[k][n] = arr[16k+n]  ->  every row of D = exact column
// sums; finish 16->1 with wave32 shuffles. EXEC all-1s (32 threads, no
// divergence before WMMA) as the ISA requires.
// ---------------------------------------------------------------------------
__device__ __forceinline__ float sum64(const float* __restrict__ arr, int lane) {
#if __has_builtin(__builtin_amdgcn_wmma_f32_16x16x4_f32)
    v2f a; a.x = 1.0f; a.y = 1.0f;
    v2f bv; bv.x = arr[lane]; bv.y = arr[lane + 32];   // B rows 0-1 / 2-3
    v8f c = {};
    v8f d = __builtin_amdgcn_wmma_f32_16x16x4_f32(
        false, a, false, bv, (short)0, c, false, false);
    float p = d[0];                 // lanes 0..15 hold D[0][n] = column sums
    p += __shfl_xor(p, 8, 32);
    p += __shfl_xor(p, 4, 32);
    p += __shfl_xor(p, 2, 32);
    p += __shfl_xor(p, 1, 32);
    return __shfl(p, 0, 32);
#else
    float p = arr[lane] + arr[lane + 32];
    for (int mlane = 16; mlane >= 1; mlane >>= 1) p += __shfl_xor(p, mlane, 32);
    return p;
#endif
}

__global__ __launch_bounds__(32) void k_final(
    const float* __restrict__ pb_posf, const float* __restrict__ pb_l1,
    const float* __restrict__ pb_cep,  const float* __restrict__ pb_negsum,
    const float* __restrict__ pb_negcnt, float* __restrict__ out)
{
    const int lane = threadIdx.x;
    float Spos = sum64(pb_posf,   lane);
    float Sl1  = sum64(pb_l1,     lane);
    float Scep = sum64(pb_cep,    lane);
    float Sns  = sum64(pb_negsum, lane);
    float Snc  = sum64(pb_negcnt, lane);
    if (lane == 0) {
        float loss_l = (Spos > 0.0f) ? Sl1 / (4.0f * Spos) : 0.0f;
        float lcp    = (Spos > 0.0f) ? Scep / Spos : 0.0f;
        float lcn    = (Snc  > 0.0f) ? Sns / Snc : 0.0f;
        out[0] = loss_l;
        out[1] = lcp + lcn;
        out[2] = lcp;
        out[3] = lcn;
    }
}

// ---------------------------------------------------------------------------
extern "C" void kernel_launch(void* const* d_in, const int* in_sizes, int n_in,
                              void* d_out, int out_size, void* d_ws, size_t ws_size,
                              hipStream_t stream) {
    (void)in_sizes; (void)n_in; (void)out_size; (void)ws_size;
    const float* loc  = (const float*)d_in[0];   // [64][8732][4]
    const float* conf = (const float*)d_in[1];   // [64][8732][2]
    const float* dbox = (const float*)d_in[2];   // [8732][4]
    const float* tgt  = (const float*)d_in[3];   // [64][50][5]
    float* out = (float*)d_out;

    char* ws = (char*)d_ws;
    float* mined    = (float*)ws; ws += (size_t)NB * ND * sizeof(float);
    int*   pb_npos  = (int*)  ws; ws += NB * sizeof(int);
    float* pb_posf  = (float*)ws; ws += NB * sizeof(float);
    float* pb_l1    = (float*)ws; ws += NB * sizeof(float);
    float* pb_cep   = (float*)ws; ws += NB * sizeof(float);
    float* pb_nsum  = (float*)ws; ws += NB * sizeof(float);
    float* pb_ncnt  = (float*)ws; ws += NB * sizeof(float);

    k_match<<<NB, THREADS, 0, stream>>>(loc, conf, dbox, tgt, mined,
                                        pb_npos, pb_posf, pb_l1, pb_cep);
    k_mine <<<NB, THREADS, 0, stream>>>(mined, pb_npos, pb_nsum, pb_ncnt);
    k_final<<<1, 32, 0, stream>>>(pb_posf, pb_l1, pb_cep, pb_nsum, pb_ncnt, out);
}